// H2GCNConv_7129645711842
// MI455X (gfx1250) — compile-verified
//
#include <hip/hip_runtime.h>
#include <hip/hip_bf16.h>

// ---------------------------------------------------------------------------
// H2GCN dual-SpMM (segment-sum) for MI455X / gfx1250.
//   out[:, 0:128]   = segment_sum(w1 * x[col1], row1)
//   out[:, 128:256] = segment_sum(w2 * x[col2], row2)
//
// Memory-bound (AI ~ 0.25 FLOP/B): x (25.6MB), out (51.2MB) and the edge
// arrays (~48MB) all fit in the 192MB global L2, so random gathers/atomics
// resolve at L2 speed. One wave32 per edge: 32 lanes x float4 = the full
// 128-wide feature row. Edge metadata is staged LDS-side via CDNA5 async
// copies (global_load_async_to_lds_*, ASYNCcnt) so the wave's VMEM slots are
// spent on the 512B x-row gather + 512B atomic burst, not skinny index loads.
// All row/col offset math is 32-bit so loads/atomics use SGPR-base +
// 32-bit-VGPR-offset addressing.
// ---------------------------------------------------------------------------

#define WAVES_PER_BLOCK 8
#define EDGES_PER_WAVE  32
#define EDGES_PER_BLOCK (WAVES_PER_BLOCK * EDGES_PER_WAVE)
#define D               128   // feature width
#define OUT_STRIDE      256   // concatenated [N, 2d]

__global__ __launch_bounds__(256)
void h2gcn_zero_kernel(float4* __restrict__ out, long long n4)
{
    long long i      = (long long)blockIdx.x * blockDim.x + threadIdx.x;
    long long stride = (long long)gridDim.x * blockDim.x;
    float4 z; z.x = 0.f; z.y = 0.f; z.z = 0.f; z.w = 0.f;
    for (; i < n4; i += stride) out[i] = z;
}

__global__ __launch_bounds__(256)
void h2gcn_spmm_scatter_kernel(const long long* __restrict__ rows,
                               const long long* __restrict__ cols,
                               const float*     __restrict__ ew,
                               const float*     __restrict__ x,
                               float*           __restrict__ out,
                               long long E, int outOff)
{
    // Per-wave staging area for 32 edges of metadata (no cross-wave sharing,
    // so only s_wait_asynccnt is needed -- no workgroup barrier).
    __shared__ long long s_row[WAVES_PER_BLOCK][EDGES_PER_WAVE];
    __shared__ long long s_col[WAVES_PER_BLOCK][EDGES_PER_WAVE];
    __shared__ float     s_w  [WAVES_PER_BLOCK][EDGES_PER_WAVE];

    const int lane = threadIdx.x & 31;
    const int wave = threadIdx.x >> 5;

    const long long base =
        (long long)blockIdx.x * EDGES_PER_BLOCK + (long long)wave * EDGES_PER_WAVE;
    long long rem = E - base;
    if (rem <= 0) return;                       // wave-uniform exit
    const int n = rem < EDGES_PER_WAVE ? (int)rem : EDGES_PER_WAVE;

    // --- Stage this wave's 32 edges of (row, col, w) into LDS via the CDNA5
    // --- async copy path (VGLOBAL -> LDS DMA, tracked by ASYNCcnt).
    if (lane < n) {
        const long long e = base + lane;
        // Low 32 bits of a generic LDS pointer == byte offset into LDS
        // (ISA 10.2).
        unsigned lds_r = (unsigned)(size_t)(const void*)&s_row[wave][lane];
        unsigned lds_c = (unsigned)(size_t)(const void*)&s_col[wave][lane];
        unsigned lds_w = (unsigned)(size_t)(const void*)&s_w  [wave][lane];
        const void* g_r = (const void*)(rows + e);
        const void* g_c = (const void*)(cols + e);
        const void* g_w = (const void*)(ew   + e);
        asm volatile("global_load_async_to_lds_b64 %0, %1, off"
                     :: "v"(lds_r), "v"(g_r) : "memory");
        asm volatile("global_load_async_to_lds_b64 %0, %1, off"
                     :: "v"(lds_c), "v"(g_c) : "memory");
        asm volatile("global_load_async_to_lds_b32 %0, %1, off"
                     :: "v"(lds_w), "v"(g_w) : "memory");
    }
    asm volatile("s_wait_asynccnt 0x0" ::: "memory");

    // Lane-biased base pointers: per-edge addresses are then a single
    // 32-bit (row|col)*stride offset -> SGPR-base + VGPR-offset addressing.
    const float* __restrict__ xbase = x   + (unsigned)(lane * 4);
    float*       __restrict__ obase = out + (unsigned)(outOff + lane * 4);

    // --- Process the staged edges: one edge per iteration, whole wave wide.
    // lane L owns features [4L, 4L+4): one global_load_b128 of x[col] and
    // four non-returning global_atomic_add_f32 into out[row] (the wave's 32
    // lanes cover a contiguous 512B span -> clean L2 atomic burst).
    if (n == EDGES_PER_WAVE) {
        // Full tile (the common case: E1/E2 are multiples of 256).
        // Fixed trip count + unroll => deep MLP on the x-row gathers.
#pragma unroll 8
        for (int j = 0; j < EDGES_PER_WAVE; ++j) {
            const unsigned r  = (unsigned)s_row[wave][j];   // broadcast ds_load
            const unsigned c  = (unsigned)s_col[wave][j];
            const float    wj = s_w[wave][j];

            const float4 v = *(const float4*)(xbase + c * (unsigned)D);
            float* o = obase + r * (unsigned)OUT_STRIDE;
            atomicAdd(o + 0, wj * v.x);
            atomicAdd(o + 1, wj * v.y);
            atomicAdd(o + 2, wj * v.z);
            atomicAdd(o + 3, wj * v.w);
        }
    } else {
        for (int j = 0; j < n; ++j) {
            const unsigned r  = (unsigned)s_row[wave][j];
            const unsigned c  = (unsigned)s_col[wave][j];
            const float    wj = s_w[wave][j];

            const float4 v = *(const float4*)(xbase + c * (unsigned)D);
            float* o = obase + r * (unsigned)OUT_STRIDE;
            atomicAdd(o + 0, wj * v.x);
            atomicAdd(o + 1, wj * v.y);
            atomicAdd(o + 2, wj * v.z);
            atomicAdd(o + 3, wj * v.w);
        }
    }
}

extern "C" void kernel_launch(void* const* d_in, const int* in_sizes, int n_in,
                              void* d_out, int out_size, void* d_ws, size_t ws_size,
                              hipStream_t stream)
{
    (void)d_ws; (void)ws_size; (void)n_in;

    // setup_inputs() order: x, edge_index1, edge_w1, edge_index2, edge_w2
    const float*     x   = (const float*)d_in[0];
    const long long* ei1 = (const long long*)d_in[1];   // [2, E1] int64
    const float*     w1  = (const float*)d_in[2];
    const long long* ei2 = (const long long*)d_in[3];   // [2, E2] int64
    const float*     w2  = (const float*)d_in[4];
    float* out = (float*)d_out;                          // [N, 256] f32

    const long long E1 = (long long)in_sizes[1] / 2;
    const long long E2 = (long long)in_sizes[3] / 2;

    // 1) Zero the output accumulator (out_size = N * 256).
    {
        long long n4 = (long long)out_size / 4;
        int blocks = (int)((n4 + 255) / 256);
        if (blocks > 4096) blocks = 4096;
        h2gcn_zero_kernel<<<blocks, 256, 0, stream>>>((float4*)out, n4);
    }

    // 2) SpMM #1 -> columns [0,128)
    {
        int blocks = (int)((E1 + EDGES_PER_BLOCK - 1) / EDGES_PER_BLOCK);
        h2gcn_spmm_scatter_kernel<<<blocks, 256, 0, stream>>>(
            ei1, ei1 + E1, w1, x, out, E1, /*outOff=*/0);
    }

    // 3) SpMM #2 -> columns [128,256)
    {
        int blocks = (int)((E2 + EDGES_PER_BLOCK - 1) / EDGES_PER_BLOCK);
        h2gcn_spmm_scatter_kernel<<<blocks, 256, 0, stream>>>(
            ei2, ei2 + E2, w2, x, out, E2, /*outOff=*/D);
    }
}